// DupireNeuralModel_65420941853261
// MI455X (gfx1250) — compile-verified
//
#include <hip/hip_runtime.h>
#include <hip/hip_bf16.h>

// ---------------------------------------------------------------------------
// Dupire local-vol Monte Carlo path scan.
//   dW  : [N_T, M] float32 (pre-scaled by sqrt(dt))
//   out : [N_T, M] float32; row 0 = S0, row t+1 = Euler step t.
// Memory-bound streaming scan: each thread owns 4 adjacent paths (float4
// loads/stores, NT cache policy), loops over the 255 time steps.
// ---------------------------------------------------------------------------

typedef float v4f __attribute__((ext_vector_type(4)));

#define M_PATHS   262144
#define N_STEPS   256

__global__ __launch_bounds__(256) void dupire_paths_kernel(
    const float* __restrict__ dW, float* __restrict__ out) {

  constexpr float DT         = 0.004f;
  constexpr float S0         = 100.0f;
  constexpr float INV_S0     = 1.0f / 100.0f;
  constexpr float R_RATE     = 0.05f;
  constexpr float SIGMA_BASE = 0.3f;
  constexpr float X_SHIFT    = 0.1f;
  constexpr float T_SHIFT    = 0.1f;
  // jnp.linspace(0, N_T*dt, N_T): step = N_T*dt/(N_T-1), NOT dt.
  constexpr float T_STEP     = (N_STEPS * DT) / (float)(N_STEPS - 1);
  constexpr float GROWTH     = 1.0f + R_RATE * DT;   // 1 + r*dt
  constexpr float LOG2E      = 1.4426950408889634f;

  const int tid = blockIdx.x * blockDim.x + threadIdx.x;
  const long long m0 = (long long)tid * 4;           // 16B-aligned column base

  v4f S = {S0, S0, S0, S0};

  // Row 0 is the initial spot.
  __builtin_nontemporal_store(S, (v4f*)(out + m0));

  for (int t = 0; t < N_STEPS - 1; ++t) {
    // t_now + T_SHIFT (scalar per wave; SALU-friendly)
    const float tf = fmaf((float)t, T_STEP, T_SHIFT);

    const v4f dw = __builtin_nontemporal_load(
        (const v4f*)(dW + (long long)t * M_PATHS + m0));

    // Prefetch the streaming read a few rows ahead (global_prefetch_b8).
    {
      int tp = t + 8;
      if (tp > N_STEPS - 2) tp = N_STEPS - 2;
      __builtin_prefetch(dW + (long long)tp * M_PATHS + m0, 0, 1);
    }

#pragma unroll
    for (int j = 0; j < 4; ++j) {
      const float s     = S[j];
      // y = sqrt(S/S0 + x_shift) * (t + t_shift)
      const float y     = __builtin_amdgcn_sqrtf(fmaf(s, INV_S0, X_SHIFT)) * tf;
      // sigma = base + y * exp(-y)   (exp via v_exp_f32: exp2(-y*log2e))
      const float sigma = fmaf(y, __builtin_amdgcn_exp2f(-y * LOG2E), SIGMA_BASE);
      // S_new = S + r*S*dt + sigma*S*dw = S * (1 + r*dt + sigma*dw)
      S[j] = s * fmaf(sigma, dw[j], GROWTH);
    }

    __builtin_nontemporal_store(S, (v4f*)(out + (long long)(t + 1) * M_PATHS + m0));
  }
}

extern "C" void kernel_launch(void* const* d_in, const int* in_sizes, int n_in,
                              void* d_out, int out_size, void* d_ws, size_t ws_size,
                              hipStream_t stream) {
  (void)in_sizes; (void)n_in; (void)out_size; (void)d_ws; (void)ws_size;

  const float* dW = (const float*)d_in[0];
  float* out      = (float*)d_out;

  const int threads_total = M_PATHS / 4;   // 65536 threads, 4 paths each
  const int block = 256;                   // 8 wave32 per block
  const int grid  = threads_total / block; // 256 blocks

  hipLaunchKernelGGL(dupire_paths_kernel, dim3(grid), dim3(block), 0, stream,
                     dW, out);
}